// RTDETRTransformer_22789096473367
// MI455X (gfx1250) — compile-verified
//
#include <hip/hip_runtime.h>
#include <hip/hip_bf16.h>
#include <math.h>

// ---------------------------------------------------------------------------
// RT-DETR multi-scale deformable attention, fp32, WMMA 16x16x4 (gfx1250)
//   bs=16 Lq=300 Lv=8500 D=256 NH=8 NL=4 NP=4 HD=32
// Pipeline:
//   1) v     = value @ W_v + b_v     (WMMA GEMM, epilogue permute -> [b,h,pos,hd])
//   2) off   = query @ W_off + b_off (WMMA GEMM)
//   3) logit = query @ W_attn+b_attn (WMMA GEMM)
//   4) build_loc: softmax(16) + sampling locations
//   5) deform_sample: bilinear gather-accumulate (L2-resident vperm)
//   6) out   = sampled @ W_o + b_o   (WMMA GEMM)
// GEMM staging: GLOBAL_LOAD_ASYNC_TO_LDS_B128 double buffering (ASYNCcnt).
// ---------------------------------------------------------------------------

typedef float v2f __attribute__((ext_vector_type(2)));
typedef float v8f __attribute__((ext_vector_type(8)));

#define BS   16
#define LQ   300
#define LV   8500
#define DIM  256
#define NH   8
#define NL   4
#define NP   4
#define HD   32

#define KDIM 256   // all GEMMs have K = 256
#define BM   32
#define BN   64
#define KB   64
#define NCH  (KDIM / KB)   // 4 K-chunks
#define AST  (KB + 4)      // A LDS row stride: 272B keeps B128 align, de-banks

#if defined(__gfx1250__) && __has_builtin(__builtin_amdgcn_global_load_async_to_lds_b128)
#define USE_ASYNC_LDS 1
#else
#define USE_ASYNC_LDS 0
#endif

#if defined(__gfx1250__) && __has_builtin(__builtin_amdgcn_s_wait_asynccnt)
#define ASYNC_WAIT(n) __builtin_amdgcn_s_wait_asynccnt(n)
#else
#define ASYNC_WAIT(n) asm volatile("s_wait_asynccnt %0" :: "n"(n) : "memory")
#endif

// pointer types matching the async-to-LDS builtin's parameters:
//   src: global (AS1) pointer to 16B int vector; dst: LDS (AS3) same pointee
typedef int v4i_b128 __attribute__((vector_size(16)));
typedef __attribute__((address_space(1))) v4i_b128 gas_b128;
typedef __attribute__((address_space(3))) v4i_b128 las_b128;

// ---------------------------------------------------------------------------
// Generic fp32 WMMA GEMM: C[M,N] = A[M,K=256] @ B[256,N] + bias[N]
// permuteForValue: scatter output into [b, h, pos, hd] layout instead.
// Requires M % 32 == 0, N % 64 == 0 (true for all call sites).
// ---------------------------------------------------------------------------
__global__ __launch_bounds__(256) void gemm_wmma_f32(
    const float* __restrict__ A, const float* __restrict__ B,
    const float* __restrict__ bias, float* __restrict__ C,
    int M, int N, int permuteForValue)
{
    __shared__ float As[2][BM][AST];
    __shared__ float Bs[2][KB][BN];

    const int tid  = threadIdx.x;
    const int lane = tid & 31;
    const int wave = tid >> 5;
    const int half = lane >> 4;     // 0: K0/K1 & rows M, 1: K2/K3 & rows M+8
    const int lm   = lane & 15;
    const int wm   = wave >> 2;     // 0..1  (M sub-tile)
    const int wn   = wave & 3;      // 0..3  (N sub-tile)
    const int rowBase = blockIdx.x * BM;
    const int colBase = blockIdx.y * BN;

    // stage one BMxKB A panel + KBxBN B panel into buffer `buf`
    auto stage = [&](int buf, int kb) {
#if USE_ASYNC_LDS
        #pragma unroll
        for (int it = 0; it < 2; ++it) {            // A: 512 B128, 2/thread
            int f  = tid + it * 256;
            int r  = f >> 4;
            int c4 = (f & 15) << 2;
            __builtin_amdgcn_global_load_async_to_lds_b128(
                (gas_b128*)(A + (size_t)(rowBase + r) * KDIM + kb + c4),
                (las_b128*)&As[buf][r][c4], 0, 0);
        }
        #pragma unroll
        for (int it = 0; it < 4; ++it) {            // B: 1024 B128, 4/thread
            int f  = tid + it * 256;
            int r  = f >> 4;
            int c4 = (f & 15) << 2;
            __builtin_amdgcn_global_load_async_to_lds_b128(
                (gas_b128*)(B + (size_t)(kb + r) * N + colBase + c4),
                (las_b128*)&Bs[buf][r][c4], 0, 0);
        }
#else
        #pragma unroll
        for (int it = 0; it < 2; ++it) {
            int f  = tid + it * 256;
            int r  = f >> 4;
            int c4 = (f & 15) << 2;
            float4 v = *(const float4*)(A + (size_t)(rowBase + r) * KDIM + kb + c4);
            *(float4*)(&As[buf][r][c4]) = v;
        }
        #pragma unroll
        for (int it = 0; it < 4; ++it) {
            int f  = tid + it * 256;
            int r  = f >> 4;
            int c4 = (f & 15) << 2;
            float4 v = *(const float4*)(B + (size_t)(kb + r) * N + colBase + c4);
            *(float4*)(&Bs[buf][r][c4]) = v;
        }
#endif
    };

    v8f acc = {};

    stage(0, 0);
    #pragma unroll
    for (int c = 0; c < NCH; ++c) {
        if (c + 1 < NCH) stage((c + 1) & 1, (c + 1) * KB);   // prefetch next chunk
#if USE_ASYNC_LDS
        if (c + 1 < NCH) ASYNC_WAIT(6);   // chunk c landed; c+1 still in flight
        else             ASYNC_WAIT(0);
#endif
        __syncthreads();

        const int buf = c & 1;
        // 16 x V_WMMA_F32_16X16X4_F32 per K-chunk
        #pragma unroll
        for (int kk = 0; kk < KB; kk += 4) {
            const int k2 = kk + 2 * half;
            v2f a, b;
            // A 16x4 frag: half 0 -> (K0,K1), half 1 -> (K2,K3), M = lm
            a.x = As[buf][wm * 16 + lm][k2];
            a.y = As[buf][wm * 16 + lm][k2 + 1];
            // B 4x16 frag: same half-wave K striping, N = lm
            b.x = Bs[buf][k2][wn * 16 + lm];
            b.y = Bs[buf][k2 + 1][wn * 16 + lm];
            acc = __builtin_amdgcn_wmma_f32_16x16x4_f32(
                      false, a, false, b, (short)0, acc, false, false);
        }
        __syncthreads();
    }

    // epilogue: C/D layout -> VGPR r holds M = r + 8*half, N = lm
    const int col = colBase + wn * 16 + lm;
    const float bv = bias[col];
    #pragma unroll
    for (int r = 0; r < 8; ++r) {
        const int row = rowBase + wm * 16 + r + 8 * half;
        const float val = acc[r] + bv;
        if (!permuteForValue) {
            C[(size_t)row * N + col] = val;
        } else {
            // row = b*Lv + pos ; col = h*HD + d  ->  [b][h][pos][d]
            const int b_ = row / LV;
            const int i  = row - b_ * LV;
            const int h  = col >> 5;
            const int d  = col & 31;
            C[(((size_t)(b_ * NH + h)) * LV + i) * HD + d] = val;
        }
    }
}

// ---------------------------------------------------------------------------
// Softmax over 16 (NL*NP) logits per head + sampling-location computation.
// One thread per (b,q,h)  -> 38400 threads.
// ---------------------------------------------------------------------------
__global__ __launch_bounds__(256) void build_loc(
    const float* __restrict__ off,     // [bs*Lq, 256]
    const float* __restrict__ logits,  // [bs*Lq, 128]
    const float* __restrict__ rp,      // [bs, Lq, 1, 4]
    float* __restrict__ loc,           // [gw, NL, NP, 2]
    float* __restrict__ aw)            // [gw, NL, NP]
{
    const int gw = blockIdx.x * blockDim.x + threadIdx.x;   // (b,q,h)
    if (gw >= BS * LQ * NH) return;
    const int h   = gw & (NH - 1);
    const int row = gw >> 3;            // b*Lq + q
    const float r0 = rp[row * 4 + 0];
    const float r1 = rp[row * 4 + 1];
    const float sx = rp[row * 4 + 2] * 0.125f;   // / NP * 0.5
    const float sy = rp[row * 4 + 3] * 0.125f;

    const float* lg = logits + (size_t)row * (NH * NL * NP) + h * (NL * NP);
    float mx = lg[0];
    #pragma unroll
    for (int j = 1; j < NL * NP; ++j) mx = fmaxf(mx, lg[j]);
    float e[NL * NP];
    float s = 0.f;
    #pragma unroll
    for (int j = 0; j < NL * NP; ++j) { e[j] = __expf(lg[j] - mx); s += e[j]; }
    const float inv = 1.f / s;

    const float* ob = off + (size_t)row * (NH * NL * NP * 2) + h * (NL * NP * 2);
    #pragma unroll
    for (int j = 0; j < NL * NP; ++j) {
        aw[(size_t)gw * (NL * NP) + j] = e[j] * inv;
        loc[((size_t)gw * (NL * NP) + j) * 2 + 0] = r0 + ob[2 * j + 0] * sx;
        loc[((size_t)gw * (NL * NP) + j) * 2 + 1] = r1 + ob[2 * j + 1] * sy;
    }
}

// ---------------------------------------------------------------------------
// Deformable bilinear sampling. One wave32 per (b,q,h); lane = channel d.
// Each corner fetch is 32 consecutive floats -> one coalesced 128B line.
// vperm (139 MB) fits in the 192 MB L2, so phase-1 prefetches make the
// phase-2 gathers L2 hits with latency already hidden.
// ---------------------------------------------------------------------------
__global__ __launch_bounds__(256) void deform_sample(
    const float* __restrict__ vperm,   // [bs, NH, Lv, HD]
    const float* __restrict__ loc,     // [gw, NL, NP, 2]
    const float* __restrict__ aw,      // [gw, NL, NP]
    float* __restrict__ sampled)       // [bs, Lq, 256]
{
    const int gw = blockIdx.x * 8 + (threadIdx.x >> 5);  // 0..38399
    const int d  = threadIdx.x & 31;
    const int h  = gw & (NH - 1);
    const int row = gw >> 3;            // b*Lq + q
    const int b_  = row / LQ;

    const float* base = vperm + ((size_t)(b_ * NH + h)) * LV * HD;

    // phase 1: load point parameters, issue anchor-corner prefetches
    float w[NL * NP], wx[NL * NP], wy[NL * NP];
    int   x0a[NL * NP], y0a[NL * NP];
    #pragma unroll
    for (int l = 0; l < NL; ++l) {
        const int Hs = (l == 0) ? 80 : (l == 1) ? 40 : (l == 2) ? 20 : 10;
        const int st = (l == 0) ? 0 : (l == 1) ? 6400 : (l == 2) ? 8000 : 8400;
        const float fH = (float)Hs;
        #pragma unroll
        for (int p = 0; p < NP; ++p) {
            const int j = l * NP + p;
            w[j] = aw[(size_t)gw * (NL * NP) + j];
            const float x = loc[((size_t)gw * (NL * NP) + j) * 2 + 0] * fH - 0.5f;
            const float y = loc[((size_t)gw * (NL * NP) + j) * 2 + 1] * fH - 0.5f;
            const float x0f = floorf(x), y0f = floorf(y);
            wx[j] = x - x0f; wy[j] = y - y0f;
            x0a[j] = (int)x0f; y0a[j] = (int)y0f;
            const int xc = min(max(x0a[j], 0), Hs - 1);
            const int yc = min(max(y0a[j], 0), Hs - 1);
            __builtin_prefetch(base + (size_t)(st + yc * Hs + xc) * HD + d, 0, 3);
        }
    }

    // phase 2: bilinear accumulate
    float acc = 0.f;
    #pragma unroll
    for (int l = 0; l < NL; ++l) {
        const int Hs = (l == 0) ? 80 : (l == 1) ? 40 : (l == 2) ? 20 : 10;
        const int st = (l == 0) ? 0 : (l == 1) ? 6400 : (l == 2) ? 8000 : 8400;
        #pragma unroll
        for (int p = 0; p < NP; ++p) {
            const int j = l * NP + p;
            #pragma unroll
            for (int c = 0; c < 4; ++c) {
                const int dx = c & 1, dy = c >> 1;
                const int xi = x0a[j] + dx, yi = y0a[j] + dy;
                const float cw = w[j] * (dx ? wx[j] : 1.f - wx[j])
                                      * (dy ? wy[j] : 1.f - wy[j]);
                if (xi >= 0 && xi < Hs && yi >= 0 && yi < Hs) {
                    acc += cw * base[(size_t)(st + yi * Hs + xi) * HD + d];
                }
            }
        }
    }
    sampled[((size_t)row) * DIM + h * HD + d] = acc;
}

// ---------------------------------------------------------------------------
extern "C" void kernel_launch(void* const* d_in, const int* in_sizes, int n_in,
                              void* d_out, int out_size, void* d_ws, size_t ws_size,
                              hipStream_t stream) {
    (void)in_sizes; (void)n_in; (void)out_size; (void)ws_size;

    const float* query  = (const float*)d_in[0];   // [16,300,256]
    const float* rp     = (const float*)d_in[1];   // [16,300,1,4]
    const float* value  = (const float*)d_in[2];   // [16,8500,256]
    const float* W_off  = (const float*)d_in[3];   // [256,256]
    const float* b_off  = (const float*)d_in[4];
    const float* W_attn = (const float*)d_in[5];   // [256,128]
    const float* b_attn = (const float*)d_in[6];
    const float* W_v    = (const float*)d_in[7];   // [256,256]
    const float* b_v    = (const float*)d_in[8];
    const float* W_o    = (const float*)d_in[9];   // [256,256]
    const float* b_o    = (const float*)d_in[10];
    float* out = (float*)d_out;

    // workspace partition (floats)
    float* ws = (float*)d_ws;
    float* ws_vperm = ws;                                     // 34,816,000
    float* ws_off   = ws_vperm + (size_t)BS * NH * LV * HD;   // 4800*256
    float* ws_log   = ws_off   + (size_t)BS * LQ * 256;       // 4800*128
    float* ws_loc   = ws_log   + (size_t)BS * LQ * 128;       // 38400*32
    float* ws_aw    = ws_loc   + (size_t)BS * LQ * NH * NL * NP * 2;  // 38400*16
    float* ws_samp  = ws_aw    + (size_t)BS * LQ * NH * NL * NP;      // 4800*256

    const int Mq = BS * LQ;        // 4800
    const int Mv = BS * LV;        // 136000

    // 1) value projection -> head-major permuted layout
    gemm_wmma_f32<<<dim3(Mv / BM, DIM / BN), 256, 0, stream>>>(
        value, W_v, b_v, ws_vperm, Mv, DIM, 1);
    // 2) sampling offsets
    gemm_wmma_f32<<<dim3(Mq / BM, 256 / BN), 256, 0, stream>>>(
        query, W_off, b_off, ws_off, Mq, 256, 0);
    // 3) attention logits
    gemm_wmma_f32<<<dim3(Mq / BM, 128 / BN), 256, 0, stream>>>(
        query, W_attn, b_attn, ws_log, Mq, 128, 0);
    // 4) softmax + locations
    build_loc<<<(BS * LQ * NH + 255) / 256, 256, 0, stream>>>(
        ws_off, ws_log, rp, ws_loc, ws_aw);
    // 5) bilinear gather-accumulate
    deform_sample<<<(BS * LQ * NH) / 8, 256, 0, stream>>>(
        ws_vperm, ws_loc, ws_aw, ws_samp);
    // 6) output projection
    gemm_wmma_f32<<<dim3(Mq / BM, DIM / BN), 256, 0, stream>>>(
        ws_samp, W_o, b_o, out, Mq, DIM, 0);
}